// GraphMaxComponentRelativeError_40346922778982
// MI455X (gfx1250) — compile-verified
//
#include <hip/hip_runtime.h>
#include <stdint.h>

typedef uint32_t u32;
typedef uint64_t u64;
typedef unsigned long long ull;

#define NCOMP      8
#define TILE_NODES 512          // 512 rows * 32B = 16KB per tile buffer
#define BLK        256
#define NB         2048
#define GMAX       64           // max graphs spanned by one block's chunk
#define GCAP       1024         // global table capacity (graphs)

#if defined(__gfx1250__) && __has_builtin(__builtin_amdgcn_tensor_load_to_lds) && __has_builtin(__builtin_amdgcn_s_wait_tensorcnt)
#define HAVE_TDM 1
#else
#define HAVE_TDM 0
#endif

#if HAVE_TDM
typedef uint32_t v4u __attribute__((ext_vector_type(4)));
typedef int      v8i __attribute__((ext_vector_type(8)));
typedef int      v4i __attribute__((ext_vector_type(4)));

// Issue one TDM DMA: tile_elems f32 elements from gsrc to the LDS byte address in lds_byte.
// 1-D tile (tile_dim1=1). td0 = tensor_dim0 (elements remaining from tile
// start to end of tensor) so hardware OOB returns zeros past the end.
__device__ inline void tdm_load_tile(const float* gsrc, u32 lds_byte,
                                     u32 tile_elems, u32 td0) {
  u64 ga = (u64)(uintptr_t)gsrc;
  v4u g0 = { 1u,                                     // count=1 (valid), user mode
             lds_byte,                               // lds_addr [63:32]
             (u32)ga,                                // global_addr[31:0]
             (u32)((ga >> 32) & 0x01ffffffu) | 0x80000000u }; // ga[56:32] | type=2
  v8i g1 = { (int)0x00020000u,                       // data_size=2 (4B), mask=0
             (int)((td0 & 0xffffu) << 16),           // tensor_dim0[15:0] @ [63:48]
             (int)(((td0 >> 16) & 0xffffu) | (1u << 16)), // td0[31:16] | tensor_dim1=1
             (int)((tile_elems & 0xffffu) << 16),    // tile_dim0 @ [127:112]
             (int)1,                                 // tile_dim1=1, tile_dim2=0
             (int)td0,                               // tensor_dim0_stride[31:0]
             0, 0 };
  v4i gz4 = { 0, 0, 0, 0 };
  v8i gz8 = { 0, 0, 0, 0, 0, 0, 0, 0 };
  __builtin_amdgcn_tensor_load_to_lds(g0, g1, gz4, gz4, gz8, 0);
}
#endif

__device__ inline u64 packkey(float a, u32 inv) {
  return ((u64)__float_as_uint(a) << 32) | (u64)inv;
}

__global__ void k_init(ull* gtable, int nEntries) {
  int i = blockIdx.x * blockDim.x + threadIdx.x;
  if (i < nEntries) gtable[i] = 0ull;
}

__global__ __launch_bounds__(BLK)
void k_main(const float* __restrict__ target, const int* __restrict__ batch,
            ull* __restrict__ gtable, int n) {
  __shared__ float tbuf[2][TILE_NODES * NCOMP];     // 32 KB double buffer
  __shared__ ull   table[GMAX * NCOMP];             // 4 KB per-block argmax table

  const int chunk = (n + (int)gridDim.x - 1) / (int)gridDim.x;
  const int s0 = (int)blockIdx.x * chunk;
  if (s0 >= n) return;
  int s1 = s0 + chunk; if (s1 > n) s1 = n;

  for (int e = threadIdx.x; e < GMAX * NCOMP; e += BLK) table[e] = 0ull;

  const int  gBase = batch[s0];
  const int  gLast = batch[s1 - 1];
  const bool fits  = (gLast - gBase) < GMAX;
  __syncthreads();

  auto flushf = [&](int g, const u64* m) {
    if (fits) {
      int off = (g - gBase) * NCOMP;
      #pragma unroll
      for (int c = 0; c < NCOMP; ++c)
        if (m[c]) atomicMax(&table[off + c], (ull)m[c]);
    } else if (g < GCAP) {
      #pragma unroll
      for (int c = 0; c < NCOMP; ++c)
        if (m[c]) atomicMax(&gtable[g * NCOMP + c], (ull)m[c]);
    }
  };

  const int numTiles = (s1 - s0 + TILE_NODES - 1) / TILE_NODES;
  u64 m[NCOMP];
  #pragma unroll
  for (int c = 0; c < NCOMP; ++c) m[c] = 0;
  int curG = -1;

#if HAVE_TDM
  const bool issuer = (threadIdx.x < 32);  // wave 0 drives the DMA pipeline
  if (issuer) {
    int rows0 = s1 - s0; if (rows0 > TILE_NODES) rows0 = TILE_NODES;
    tdm_load_tile(target + (size_t)s0 * NCOMP, (u32)(uintptr_t)(void*)tbuf[0],
                  (u32)(rows0 * NCOMP), (u32)((n - s0) * NCOMP));
  }
#endif

  for (int t = 0; t < numTiles; ++t) {
    float* buf = tbuf[t & 1];
    const int base = s0 + t * TILE_NODES;
    int rows = s1 - base; if (rows > TILE_NODES) rows = TILE_NODES;

#if HAVE_TDM
    if (issuer) {
      if (t + 1 < numTiles) {
        const int nb = s0 + (t + 1) * TILE_NODES;
        int nrows = s1 - nb; if (nrows > TILE_NODES) nrows = TILE_NODES;
        tdm_load_tile(target + (size_t)nb * NCOMP,
                      (u32)(uintptr_t)(void*)tbuf[(t + 1) & 1],
                      (u32)(nrows * NCOMP), (u32)((n - nb) * NCOMP));
        __builtin_amdgcn_s_wait_tensorcnt(1);   // tile t landed in LDS
      } else {
        __builtin_amdgcn_s_wait_tensorcnt(0);
      }
    }
    __syncthreads();                            // release consumers
#else
    for (int r = threadIdx.x; r < rows * 2; r += BLK)
      ((float4*)buf)[r] = ((const float4*)(target + (size_t)base * NCOMP))[r];
    __syncthreads();
#endif

    for (int lo = threadIdx.x; lo < rows; lo += BLK) {
      const int i = base + lo;
      const int g = batch[i];
      if (g != curG) {
        if (curG >= 0) flushf(curG, m);
        curG = g;
        #pragma unroll
        for (int c = 0; c < NCOMP; ++c) m[c] = 0;
      }
      const float4 r0 = ((const float4*)buf)[lo * 2 + 0];
      const float4 r1 = ((const float4*)buf)[lo * 2 + 1];
      const u32 inv = ~(u32)i;                  // max key => min index tie-break
      u64 k;
      k = packkey(fabsf(r0.x), inv); if (k > m[0]) m[0] = k;
      k = packkey(fabsf(r0.y), inv); if (k > m[1]) m[1] = k;
      k = packkey(fabsf(r0.z), inv); if (k > m[2]) m[2] = k;
      k = packkey(fabsf(r0.w), inv); if (k > m[3]) m[3] = k;
      k = packkey(fabsf(r1.x), inv); if (k > m[4]) m[4] = k;
      k = packkey(fabsf(r1.y), inv); if (k > m[5]) m[5] = k;
      k = packkey(fabsf(r1.z), inv); if (k > m[6]) m[6] = k;
      k = packkey(fabsf(r1.w), inv); if (k > m[7]) m[7] = k;
    }
    __syncthreads();   // everyone done with buf before it is re-DMA'd
  }
  if (curG >= 0) flushf(curG, m);
  __syncthreads();

  if (fits) {
    for (int e = threadIdx.x; e < GMAX * NCOMP; e += BLK) {
      const ull v = table[e];
      if (v) {
        const int g = gBase + e / NCOMP;
        if (g < GCAP) atomicMax(&gtable[g * NCOMP + (e % NCOMP)], v);
      }
    }
  }
}

__global__ __launch_bounds__(1024)
void k_final(const float* __restrict__ pred, const float* __restrict__ target,
             const int* __restrict__ batch, const ull* __restrict__ gtable,
             float* __restrict__ out, int n) {
  __shared__ float red[32];
  const int G = batch[n - 1] + 1;               // num_graphs = batch.max()+1
  const int total = G * NCOMP;
  float s = 0.f;
  for (int e = threadIdx.x; e < total; e += blockDim.x) {
    const ull v = gtable[e];
    const int idx = v ? (int)~(u32)v : (n - 1); // empty segment -> n-1 (ref guard)
    const int c = e % NCOMP;
    const float t = target[(size_t)idx * NCOMP + c];
    const float p = pred[(size_t)idx * NCOMP + c];
    s += fabsf(p - t) / (fabsf(t) + 1e-8f);
  }
  for (int off = 16; off; off >>= 1) s += __shfl_down(s, off, 32);
  const int wid = threadIdx.x >> 5;
  if ((threadIdx.x & 31) == 0) red[wid] = s;
  __syncthreads();
  if (threadIdx.x < 32) {
    const int nw = blockDim.x >> 5;
    float v = (threadIdx.x < nw) ? red[threadIdx.x] : 0.f;
    for (int off = 16; off; off >>= 1) v += __shfl_down(v, off, 32);
    if (threadIdx.x == 0) out[0] = v * 10000.0f / (float)total;
  }
}

extern "C" void kernel_launch(void* const* d_in, const int* in_sizes, int n_in,
                              void* d_out, int out_size, void* d_ws, size_t ws_size,
                              hipStream_t stream) {
  const float* pred   = (const float*)d_in[0];
  const float* target = (const float*)d_in[1];
  const int*   batch  = (const int*)d_in[2];
  // d_in[3] = x : unused by the reference computation
  const int n = in_sizes[2];

  ull* gtable = (ull*)d_ws;                     // 1024*8*8B = 64KB scratch
  const int entries = GCAP * NCOMP;
  k_init<<<(entries + 255) / 256, 256, 0, stream>>>(gtable, entries);
  k_main<<<NB, BLK, 0, stream>>>(target, batch, gtable, n);
  k_final<<<1, 1024, 0, stream>>>(pred, target, batch, gtable, (float*)d_out, n);
}